// EmbracementLayer_63230508532019
// MI455X (gfx1250) — compile-verified
//
#include <hip/hip_runtime.h>

// Reference shapes (fixed by the harness):
//   tokens : [BS, SEQ, EMB] float32   = [64, 512, 1024]
//   idx    : [BS, EMB]      int32     = [64, 1024]
//   out    : [BS, EMB]      float32   = [64, 1024]
// out[b, j] = tokens[b, idx[b, j], j]
//
// 0-FLOP latency-bound gather. This revision forces the cheap addressing form:
//  - b comes from blockIdx.x  -> batch base pointers are SCALAR (SGPR pair),
//    so every access is global_load/store with SADDR + 32-bit VGPR offset
//    (GVS mode), no per-lane 64-bit address arithmetic.
//  - all per-lane offsets are uint32 element offsets (max 2 MB byte offset,
//    zero-extended, no v_ashrrev/v_add_nc_u64 chains).
//  - 4 consecutive j per lane: B128 idx load + B128 store (coalesced),
//    4 independent scattered NT b32 gathers outstanding per lane (MLP).
//  - 128 workgroups x 128 threads -> wide WGP spread for latency hiding.

#define BS_   64
#define SEQ_  512
#define EMB_  1024

__global__ __launch_bounds__(128) void embrace_gather_kernel(
    const float* __restrict__ tokens,
    const int*   __restrict__ idx,
    float*       __restrict__ out)
{
    // 2 blocks per batch row: b uniform per block (scalar), j per-lane.
    const uint32_t b = blockIdx.x >> 1;                                  // SGPR
    const uint32_t j = (((blockIdx.x & 1u) << 7) + threadIdx.x) << 2;    // VGPR, 32-bit

    // Scalar base pointers (per-block uniform) -> SADDR addressing.
    const float* __restrict__ tok_b = tokens + (size_t)b * (SEQ_ * EMB_);
    const int*   __restrict__ idx_b = idx    + (size_t)b * EMB_;
    float*       __restrict__ out_b = out    + (size_t)b * EMB_;

    // Coalesced 128-bit load of 4 sequence indices (32-bit lane offset).
    const int4 s4 = *reinterpret_cast<const int4*>(idx_b + j);

    // Unsigned 32-bit element offsets into this batch's [SEQ, EMB] slab.
    const uint32_t o0 = (uint32_t)s4.x * EMB_ + j + 0u;
    const uint32_t o1 = (uint32_t)s4.y * EMB_ + j + 1u;
    const uint32_t o2 = (uint32_t)s4.z * EMB_ + j + 2u;
    const uint32_t o3 = (uint32_t)s4.w * EMB_ + j + 3u;

    // Four independent scattered loads, all outstanding before first use.
    // NT hint: only 1/512 of the 128 MB tokens tensor is touched.
    float4 o;
    o.x = __builtin_nontemporal_load(tok_b + o0);
    o.y = __builtin_nontemporal_load(tok_b + o1);
    o.z = __builtin_nontemporal_load(tok_b + o2);
    o.w = __builtin_nontemporal_load(tok_b + o3);

    // Coalesced 128-bit store.
    *reinterpret_cast<float4*>(out_b + j) = o;
}

extern "C" void kernel_launch(void* const* d_in, const int* in_sizes, int n_in,
                              void* d_out, int out_size, void* d_ws, size_t ws_size,
                              hipStream_t stream) {
    const float* tokens = (const float*)d_in[0];  // [64, 512, 1024] f32
    const int*   idx    = (const int*)  d_in[1];  // [64, 1024]      i32
    float*       out    = (float*)      d_out;    // [64, 1024]      f32

    constexpr int grid  = BS_ * 2;  // 128 workgroups (2 per batch row)
    constexpr int block = 128;      // 4 waves per WG; each lane does 4 columns
    embrace_gather_kernel<<<grid, block, 0, stream>>>(tokens, idx, out);
}